// LSTMModelLite_1898375545334
// MI455X (gfx1250) — compile-verified
//
#include <hip/hip_runtime.h>
#include <hip/hip_bf16.h>
#include <cstdint>

// ---------------------------------------------------------------------------
// Peephole-LSTM model, MI455X (gfx1250, wave32, WMMA, async global->LDS)
//   B=128, T=512, F=784, H=256, C=10
// Phase 1: xw[B*T, 4H] = x @ kernel + bias   (WMMA f32 16x16x4, LDS-staged A,
//          double-buffered GLOBAL_LOAD_ASYNC_TO_LDS_B128 / ASYNCcnt)
// Phase 2: sequential LSTM recurrence, batch-partitioned, fused BN+tanh+FC
// ---------------------------------------------------------------------------

typedef float v2f __attribute__((ext_vector_type(2)));
typedef float v8f __attribute__((ext_vector_type(8)));
typedef int   v4i_ __attribute__((vector_size(16)));   // matches builtin param
typedef __attribute__((address_space(1))) v4i_* gv4i_p;  // global int4*
typedef __attribute__((address_space(3))) v4i_* lv4i_p;  // LDS int4*

#define NB    128
#define NT    512
#define NF    784
#define NH    256
#define N4H   1024
#define NC    10
#define NROWS (NB * NT)          // 65536
#define HPAD  260                // kernel2 h rows: 256+4 pad, conflict-free ds
#define KC    16                 // kernel1 K chunk (784 = 49 * 16)
#define NKC   (NF / KC)          // 49
#define APAD  20                 // kernel1 A row pad: 80B rows, 16B aligned

#ifndef __has_builtin
#define __has_builtin(x) 0
#endif

// ---- gfx1250 async global->LDS copy (ASYNCcnt-tracked) --------------------
__device__ __forceinline__ void async_copy_b128(const float* g, const float* l) {
#if __has_builtin(__builtin_amdgcn_global_load_async_to_lds_b128)
    __builtin_amdgcn_global_load_async_to_lds_b128(
        (gv4i_p)(uintptr_t)g,
        (lv4i_p)(uint32_t)(uintptr_t)l,   // flat low 32b == LDS byte offset
        0, 0);
#else
    const unsigned lds_off = (unsigned)(uintptr_t)l;
    asm volatile("global_load_async_to_lds_b128 %0, %1, off"
                 :: "v"(lds_off), "v"(g) : "memory");
#endif
}

#if __has_builtin(__builtin_amdgcn_s_wait_asynccnt)
#define WAIT_ASYNC(n) __builtin_amdgcn_s_wait_asynccnt(n)
#else
#define WAIT_ASYNC(n) asm volatile("s_wait_asynccnt " #n ::: "memory")
#endif

__device__ __forceinline__ v8f wmma_f32_k4(v2f a, v2f b, v8f c) {
    // D = A(16x4 f32) x B(4x16 f32) + C(16x16 f32)
    return __builtin_amdgcn_wmma_f32_16x16x4_f32(
        false, a, false, b, (short)0, c, false, false);
}

__device__ __forceinline__ float sigmoidf_(float x) {
    return 1.0f / (1.0f + __expf(-x));
}

// ---------------------------------------------------------------------------
// Kernel 1: xw = x @ kernel + bias
// Block = 256 threads (8 waves) computing 64 rows x 128 cols.
// A tile (64 x 16K) staged in LDS once per block (8x traffic reduction),
// double-buffered with async B128 copies; B streams from global per wave.
// ---------------------------------------------------------------------------
__global__ __launch_bounds__(256)
void gemm_xw_kernel(const float* __restrict__ x,
                    const float* __restrict__ w,
                    const float* __restrict__ bias,
                    float* __restrict__ xw)
{
    __shared__ float ls_a[2][64 * APAD];

    const int tid  = threadIdx.x;
    const int lane = tid & 31;
    const int wave = tid >> 5;                   // 0..7
    const int nl   = lane & 15;
    const int kh   = lane >> 4;                  // 0/1: K half
    const int n    = blockIdx.x * 128 + wave * 16 + nl;
    const int m0   = blockIdx.y * 64;

    // Async copy mapping: one float4 per thread per K chunk.
    const int arow = tid >> 2;                   // 0..63
    const int acol = (tid & 3) * 4;              // 0,4,8,12
    const float* gsrc = x + (long)(m0 + arow) * NF + acol;
    float* lrow0 = &ls_a[0][arow * APAD + acol];
    float* lrow1 = &ls_a[1][arow * APAD + acol];

    v8f acc0 = {}, acc1 = {}, acc2 = {}, acc3 = {};
    const float* wc = w + n;

    async_copy_b128(gsrc, lrow0);                // prime buffer 0

    for (int c = 0; c < NKC; ++c) {
        const int cur = c & 1;
        if (c + 1 < NKC) {                       // prefetch next chunk into other buf
            async_copy_b128(gsrc + (c + 1) * KC, cur ? lrow0 : lrow1);
            WAIT_ASYNC(1);                       // oldest (current buf) has landed
        } else {
            WAIT_ASYNC(0);
        }
        __syncthreads();                         // LDS chunk visible to all waves

        const float* wk = wc + (long)c * KC * N4H;
        const float* ac = &ls_a[cur][0];
        #pragma unroll
        for (int k = 0; k < KC; k += 4) {
            const int k0 = k + 2 * kh;
            v2f bv;                              // B: lane holds B[k0][n], B[k0+1][n]
            bv.x = wk[(long)k0 * N4H];
            bv.y = wk[(long)(k0 + 1) * N4H];
            v2f a0 = *(const v2f*)(ac + (nl +  0) * APAD + k0);
            v2f a1 = *(const v2f*)(ac + (nl + 16) * APAD + k0);
            v2f a2 = *(const v2f*)(ac + (nl + 32) * APAD + k0);
            v2f a3 = *(const v2f*)(ac + (nl + 48) * APAD + k0);
            acc0 = wmma_f32_k4(a0, bv, acc0);
            acc1 = wmma_f32_k4(a1, bv, acc1);
            acc2 = wmma_f32_k4(a2, bv, acc2);
            acc3 = wmma_f32_k4(a3, bv, acc3);
        }
        __syncthreads();                         // done reading before overwrite
    }

    const float bn = bias[n];
    #pragma unroll
    for (int r = 0; r < 8; ++r) {
        const int mr = r + 8 * kh;               // D row within tile
        xw[(long)(m0 +  0 + mr) * N4H + n] = acc0[r] + bn;
        xw[(long)(m0 + 16 + mr) * N4H + n] = acc1[r] + bn;
        xw[(long)(m0 + 32 + mr) * N4H + n] = acc2[r] + bn;
        xw[(long)(m0 + 48 + mr) * N4H + n] = acc3[r] + bn;
    }
}

// ---------------------------------------------------------------------------
// Kernel 2: sequential recurrence + fused BN/tanh/Dense.
// 8 blocks x 512 threads (16 waves). Block owns batch rows [b0, b0+16).
// Wave w's 4 accumulators are the z-tiles for gates i/f/c/o at hidden slice
// j in [16w, 16w+16)  ->  all gate math stays in registers; c-state in regs.
// h lives in LDS (padded rows) and is the WMMA A operand each step.
// ---------------------------------------------------------------------------
__global__ __launch_bounds__(512)
void lstm_kernel(const float* __restrict__ xw,
                 const float* __restrict__ R,      // rec_kernel [H,4H]
                 const float* __restrict__ w_ci,
                 const float* __restrict__ w_cf,
                 const float* __restrict__ w_co,
                 const float* __restrict__ gamma,
                 const float* __restrict__ beta,
                 const float* __restrict__ mean,
                 const float* __restrict__ var,
                 const float* __restrict__ fc_w,   // [H, C]
                 float* __restrict__ out)          // [B, T, C]
{
    __shared__ float ls_h[16 * HPAD];
    __shared__ float ls_y[16 * HPAD];
    __shared__ float ls_fc[NH * NC];

    const int tid  = threadIdx.x;
    const int lane = tid & 31;
    const int wave = tid >> 5;                    // 0..15
    const int nl   = lane & 15;
    const int kh   = lane >> 4;
    const int b0   = blockIdx.x * 16;
    const int j    = wave * 16 + nl;              // hidden index owned by lane

    // Per-lane constants (peephole + BN), invariant over t and rows.
    const float wci = w_ci[j], wcf = w_cf[j], wco = w_co[j];
    const float mn  = mean[j];
    const float rstd = rsqrtf(var[j] + 1e-3f);
    const float gm  = gamma[j], bt = beta[j];

    for (int i = tid; i < NH * NC; i += 512) ls_fc[i] = fc_w[i];
    for (int i = tid; i < 16 * HPAD; i += 512) ls_h[i] = 0.0f;

    v8f creg = {};                                // cell state, in registers

    const int m_out = tid / NC;                   // fc lanes (tid < 160)
    const int c_out = tid % NC;

    __syncthreads();

    for (int t = 0; t < NT; ++t) {
        // ---- z = xw_t + h_{t-1} @ R : init accumulators from xw ----------
        v8f acc[4];
        #pragma unroll
        for (int g = 0; g < 4; ++g) {
            const float* p = xw + ((long)b0 * NT + t) * N4H + g * NH + wave * 16 + nl;
            #pragma unroll
            for (int r = 0; r < 8; ++r) {
                const int m = r + 8 * kh;
                acc[g][r] = p[(long)m * NT * N4H];
            }
        }

        // ---- GEMM: M=16 (batch rows), N=4x16 (gate slices), K=256 --------
        #pragma unroll 4
        for (int k = 0; k < NH; k += 4) {
            const int k0 = k + 2 * kh;
            v2f a = *(const v2f*)(ls_h + nl * HPAD + k0);   // A: h rows
            #pragma unroll
            for (int g = 0; g < 4; ++g) {
                const float* rc = R + (long)g * NH + wave * 16 + nl;
                v2f bv;
                bv.x = rc[(long)k0 * N4H];
                bv.y = rc[(long)(k0 + 1) * N4H];
                acc[g] = wmma_f32_k4(a, bv, acc[g]);
            }
        }

        __syncthreads();   // everyone finished reading h_{t-1}

        // ---- gates + peepholes + BN + tanh, all in registers --------------
        #pragma unroll
        for (int r = 0; r < 8; ++r) {
            const float cp = creg[r];
            const float ig = sigmoidf_(acc[0][r] + cp * wci);
            const float fg = sigmoidf_(acc[1][r] + cp * wcf);
            const float cn = fg * cp + ig * tanhf(acc[2][r]);
            const float og = sigmoidf_(acc[3][r] + cn * wco);
            const float hn = og * tanhf(cn);
            creg[r] = cn;
            const float y = tanhf((hn - mn) * rstd * gm + bt);
            const int m = r + 8 * kh;
            ls_h[m * HPAD + j] = hn;
            ls_y[m * HPAD + j] = y;
        }

        __syncthreads();   // h_t / y_t visible

        // ---- fused Dense: logits[16,10] for this timestep -----------------
        if (tid < 16 * NC) {
            const float* yr = ls_y + m_out * HPAD;
            const float* fr = ls_fc + c_out;
            float sum = 0.0f;
            #pragma unroll 8
            for (int k = 0; k < NH; ++k)
                sum += yr[k] * fr[k * NC];
            out[((long)(b0 + m_out) * NT + t) * NC + c_out] = sum;
        }

        __syncthreads();   // fc done before ls_y is overwritten next step
    }
}

// ---------------------------------------------------------------------------
extern "C" void kernel_launch(void* const* d_in, const int* in_sizes, int n_in,
                              void* d_out, int out_size, void* d_ws, size_t ws_size,
                              hipStream_t stream) {
    const float* x      = (const float*)d_in[0];
    const float* kern   = (const float*)d_in[1];
    const float* rker   = (const float*)d_in[2];
    const float* bias   = (const float*)d_in[3];
    const float* w_ci   = (const float*)d_in[4];
    const float* w_cf   = (const float*)d_in[5];
    const float* w_co   = (const float*)d_in[6];
    const float* gamma  = (const float*)d_in[7];
    const float* beta   = (const float*)d_in[8];
    const float* mmean  = (const float*)d_in[9];
    const float* mvar   = (const float*)d_in[10];
    const float* fc_w   = (const float*)d_in[11];
    float* out = (float*)d_out;

    float* xw = (float*)d_ws;   // [B*T, 4H] = 256 MB scratch

    dim3 g1(N4H / 128, NROWS / 64);
    gemm_xw_kernel<<<g1, 256, 0, stream>>>(x, kern, bias, xw);

    lstm_kernel<<<NB / 16, 512, 0, stream>>>(xw, rker, w_ci, w_cf, w_co,
                                             gamma, beta, mmean, mvar, fc_w, out);
}